// Conv2D_79216376807984
// MI455X (gfx1250) — compile-verified
//
#include <hip/hip_runtime.h>

typedef float v2f __attribute__((ext_vector_type(2)));
typedef float v8f __attribute__((ext_vector_type(8)));

#define IMG_W   8192
#define IMG_H   8192
#define OUT_W   8186
#define OUT_H   8186
#define TILE    16
#define TILES_X 512                  // ceil(8186/16)
#define TILES_Y 512
#define N_TILES (TILES_X * TILES_Y)

// One wave32 computes one 16x16 output tile via 42 chained
// v_wmma_f32_16x16x4_f32 ops:
//   Out[m][n] = bias + sum_ky sum_i X[r0+m+ky][c0+i] * K[ky][i-n]
// The banded-Toeplitz B fragments are precomputed once per wave.
// Per tile: (1) issue all 42 A-fragment loads as one batch, (2) run the
// 42 WMMAs on two independent accumulator chains, (3) store.
__global__ __launch_bounds__(128) void conv7x7_wmma_f32(
    const float* __restrict__ X,
    const float* __restrict__ K,
    const float* __restrict__ bias,
    float* __restrict__ out)
{
    const int lane = threadIdx.x & 31;          // wave32
    const int n    = lane & 15;                 // B/C/D column index
    const int mrow = lane & 15;                 // A row index (M = lane%16)
    const int k0   = (lane >> 4) << 1;          // K sub-index: 0 (lanes 0-15) / 2 (lanes 16-31)

    const int wave_in_block = threadIdx.x >> 5;     // uniform within wave
    const int gwave  = blockIdx.x * (blockDim.x >> 5) + wave_in_block;
    const int nwaves = gridDim.x * (blockDim.x >> 5);

    const float bval = bias[0];

    // ---- Precompute banded-Toeplitz B fragments (invariant across tiles) ----
    // B_ky[i][n] = K[ky][i-n] for 0 <= i-n <= 6, else 0.   i = 4*kc + k0 + {0,1}
    v2f Bf[42];
#pragma unroll
    for (int ky = 0; ky < 7; ++ky) {
#pragma unroll
        for (int kc = 0; kc < 6; ++kc) {
            int t0 = 4 * kc + k0 - n;
            int t1 = t0 + 1;
            int c0i = t0 < 0 ? 0 : (t0 > 6 ? 6 : t0);
            int c1i = t1 < 0 ? 0 : (t1 > 6 ? 6 : t1);
            float k0v = K[ky * 7 + c0i];
            float k1v = K[ky * 7 + c1i];
            v2f b;
            b.x = (t0 >= 0 && t0 <= 6) ? k0v : 0.0f;
            b.y = (t1 >= 0 && t1 <= 6) ? k1v : 0.0f;
            Bf[ky * 6 + kc] = b;
        }
    }

    // ---- Wave-uniform grid-stride loop over 16x16 output tiles ----
    for (int tid = gwave; tid < N_TILES; tid += nwaves) {
        const int ty = tid >> 9;        // tid / TILES_X
        const int tx = tid & 511;       // tid % TILES_X
        const int r0 = ty * TILE;
        const int c0 = tx * TILE;
        const bool interior = (tx < TILES_X - 1) && (ty < TILES_Y - 1); // uniform per wave

        // -------- Stage 1: batch-load all 42 A fragments --------
        v2f Af[42];
        if (interior) {
            // Single per-lane base address; all 42 loads use immediate offsets,
            // so they issue as one clause with full memory-level parallelism.
            const float* base = X + (size_t)(r0 + mrow) * IMG_W + (size_t)(c0 + k0);
#pragma unroll
            for (int ky = 0; ky < 7; ++ky) {
#pragma unroll
                for (int kc = 0; kc < 6; ++kc) {
                    Af[ky * 6 + kc] =
                        *reinterpret_cast<const v2f*>(base + ky * IMG_W + kc * 4);
                }
            }
        } else {
            // Edge tiles: clamp reads (clamped values only feed out-of-range
            // rows/cols of the tile, which are never stored).
#pragma unroll
            for (int ky = 0; ky < 7; ++ky) {
                int row = r0 + ky + mrow;
                if (row > IMG_H - 1) row = IMG_H - 1;
                const float* rowp = X + (size_t)row * IMG_W;
#pragma unroll
                for (int kc = 0; kc < 6; ++kc) {
                    int col = c0 + 4 * kc + k0;
                    if (col > IMG_W - 2) col = IMG_W - 2;   // keep 8B load in-bounds/aligned
                    Af[ky * 6 + kc] = *reinterpret_cast<const v2f*>(rowp + col);
                }
            }
        }

        // -------- Stage 2: 42 WMMAs on two independent accumulator chains --------
        v8f acc0, acc1;
#pragma unroll
        for (int v = 0; v < 8; ++v) { acc0[v] = bval; acc1[v] = 0.0f; }

#pragma unroll
        for (int j = 0; j < 42; j += 2) {
            acc0 = __builtin_amdgcn_wmma_f32_16x16x4_f32(
                false, Af[j],     false, Bf[j],     (short)0, acc0, false, false);
            acc1 = __builtin_amdgcn_wmma_f32_16x16x4_f32(
                false, Af[j + 1], false, Bf[j + 1], (short)0, acc1, false, false);
        }

        v8f acc;
#pragma unroll
        for (int v = 0; v < 8; ++v) acc[v] = acc0[v] + acc1[v];

        // -------- Stage 3: store 16x16 tile --------
        // Lane holds (M = v + 8*(lane>=16), N = lane%16).
        const int m0 = (lane >> 4) * 8;
        const int cc = c0 + n;
        if (interior) {
            float* orow = out + (size_t)(r0 + m0) * OUT_W + cc;
#pragma unroll
            for (int v = 0; v < 8; ++v) {
                orow[(size_t)v * OUT_W] = acc[v];
            }
        } else {
            const bool colok = cc < OUT_W;
#pragma unroll
            for (int v = 0; v < 8; ++v) {
                int rr = r0 + m0 + v;
                if (colok && rr < OUT_H) {
                    out[(size_t)rr * OUT_W + cc] = acc[v];
                }
            }
        }
    }
}

extern "C" void kernel_launch(void* const* d_in, const int* in_sizes, int n_in,
                              void* d_out, int out_size, void* d_ws, size_t ws_size,
                              hipStream_t stream) {
    const float* X    = (const float*)d_in[0];   // 8192*8192 fp32
    const float* K    = (const float*)d_in[1];   // 7*7 fp32
    const float* bias = (const float*)d_in[2];   // 1 fp32
    float* out        = (float*)d_out;           // 8186*8186 fp32

    (void)in_sizes; (void)n_in; (void)out_size; (void)d_ws; (void)ws_size;

    dim3 block(128);     // 4 waves per block (wave32)
    dim3 grid(4096);     // 16384 waves -> 16 tiles per wave
    conv7x7_wmma_f32<<<grid, block, 0, stream>>>(X, K, bias, out);
}